// GCNLayer_27882927685658
// MI455X (gfx1250) — compile-verified
//
#include <hip/hip_runtime.h>
#include <hip/hip_bf16.h>

#define FEAT 128          // feature dim (fixed by reference)
#define WAVE 32           // gfx1250 is wave32

// ---------------------------------------------------------------------------
// Kernel 1: degree histograms. One thread per edge, two fire-and-forget
// f32 atomics (global_atomic_add_f32, no return -> STOREcnt only, no stall).
// ---------------------------------------------------------------------------
__global__ __launch_bounds__(256) void gcn_deg_kernel(
    const int* __restrict__ src, const int* __restrict__ dst,
    float* __restrict__ out_deg, float* __restrict__ in_deg, int E) {
  int e = blockIdx.x * blockDim.x + threadIdx.x;
  if (e < E) {
    // keep index stream ahead of use (gfx1250 global_prefetch_b8 path)
    if (e + 2048 < E) {
      __builtin_prefetch(&src[e + 2048], 0, 3);
      __builtin_prefetch(&dst[e + 2048], 0, 3);
    }
    atomicAdd(&out_deg[src[e]], 1.0f);
    atomicAdd(&in_deg[dst[e]], 1.0f);
  }
}

// ---------------------------------------------------------------------------
// Kernel 2: norms = rsqrt(max(deg, 1)). One thread per node.
// ---------------------------------------------------------------------------
__global__ __launch_bounds__(256) void gcn_norm_kernel(
    const float* __restrict__ out_deg, const float* __restrict__ in_deg,
    float* __restrict__ out_norm, float* __restrict__ in_norm, int N) {
  int i = blockIdx.x * blockDim.x + threadIdx.x;
  if (i < N) {
    out_norm[i] = rsqrtf(fmaxf(out_deg[i], 1.0f));
    in_norm[i]  = rsqrtf(fmaxf(in_deg[i], 1.0f));
  }
}

// ---------------------------------------------------------------------------
// Kernel 3: edge scatter. One wave32 per edge; lane L owns features
// [4L, 4L+4). Gathered row is one coalesced global_load_b128 per lane
// (512 B per wave). node_f (51 MB) and out (51 MB) both fit the 192 MB L2,
// so gathers and atomic RMWs are L2-resident. Atomics are non-returning.
// ---------------------------------------------------------------------------
__global__ __launch_bounds__(256) void gcn_scatter_kernel(
    const float* __restrict__ u_f, const float* __restrict__ v_f,
    const int* __restrict__ src, const int* __restrict__ dst,
    const float* __restrict__ out_norm,
    float* __restrict__ out, int E, int n_u) {
  int tid  = blockIdx.x * blockDim.x + threadIdx.x;
  int e    = tid >> 5;        // wave per edge
  int lane = tid & (WAVE - 1);
  if (e >= E) return;

  if (lane == 0 && e + 512 < E) {
    __builtin_prefetch(&src[e + 512], 0, 3);
    __builtin_prefetch(&dst[e + 512], 0, 3);
  }

  int s = src[e];            // lane-uniform; HW coalesces to one request
  int d = dst[e];
  float w = out_norm[s];     // source-side normalization, L2 hit

  const float* row = (s < n_u) ? (u_f + (size_t)s * FEAT)
                               : (v_f + (size_t)(s - n_u) * FEAT);
  float4 v = ((const float4*)row)[lane];           // global_load_b128

  float* orow = out + (size_t)d * FEAT + (size_t)lane * 4;
  atomicAdd(orow + 0, v.x * w);                    // global_atomic_add_f32 x4
  atomicAdd(orow + 1, v.y * w);
  atomicAdd(orow + 2, v.z * w);
  atomicAdd(orow + 3, v.w * w);
}

// ---------------------------------------------------------------------------
// Kernel 4: destination-side normalization: out[i,:] *= in_norm[i].
// One wave per row, float4 per lane (b128 load + b128 store).
// ---------------------------------------------------------------------------
__global__ __launch_bounds__(256) void gcn_scale_kernel(
    float* __restrict__ out, const float* __restrict__ in_norm, int N) {
  int tid  = blockIdx.x * blockDim.x + threadIdx.x;
  int i    = tid >> 5;
  int lane = tid & (WAVE - 1);
  if (i >= N) return;
  float w = in_norm[i];
  float4* p = (float4*)(out + (size_t)i * FEAT) + lane;
  float4 v = *p;
  v.x *= w; v.y *= w; v.z *= w; v.w *= w;
  *p = v;
}

// ---------------------------------------------------------------------------
// Launch. Inputs (reference order): u_f f32[n_u,128], v_f f32[n_v,128],
// src i32[E], dst i32[E]. Output: f32[n_u+n_v, 128].
// Workspace: out_deg | in_deg | out_norm | in_norm (4 * N floats = 1.6 MB).
// Graph-capture safe: only hipMemsetAsync + kernel launches on `stream`.
// ---------------------------------------------------------------------------
extern "C" void kernel_launch(void* const* d_in, const int* in_sizes, int n_in,
                              void* d_out, int out_size, void* d_ws, size_t ws_size,
                              hipStream_t stream) {
  const float* u_f = (const float*)d_in[0];
  const float* v_f = (const float*)d_in[1];
  const int*   src = (const int*)d_in[2];
  const int*   dst = (const int*)d_in[3];

  const int n_u = in_sizes[0] / FEAT;
  const int n_v = in_sizes[1] / FEAT;
  const int N   = n_u + n_v;
  const int E   = in_sizes[2];

  float* out      = (float*)d_out;
  float* out_deg  = (float*)d_ws;
  float* in_deg   = out_deg + N;
  float* out_norm = in_deg + N;
  float* in_norm  = out_norm + N;

  // zero accumulation targets every call (harness does not re-poison)
  hipMemsetAsync(out, 0, (size_t)N * FEAT * sizeof(float), stream);
  hipMemsetAsync(out_deg, 0, (size_t)2 * N * sizeof(float), stream);

  const int B = 256;

  gcn_deg_kernel<<<(E + B - 1) / B, B, 0, stream>>>(src, dst, out_deg, in_deg, E);

  gcn_norm_kernel<<<(N + B - 1) / B, B, 0, stream>>>(out_deg, in_deg,
                                                     out_norm, in_norm, N);

  long long scatter_threads = (long long)E * WAVE;
  gcn_scatter_kernel<<<(unsigned)((scatter_threads + B - 1) / B), B, 0, stream>>>(
      u_f, v_f, src, dst, out_norm, out, E, n_u);

  long long scale_threads = (long long)N * WAVE;
  gcn_scale_kernel<<<(unsigned)((scale_threads + B - 1) / B), B, 0, stream>>>(
      out, in_norm, N);
}